// DeformableConv2D_8753143349877
// MI455X (gfx1250) — compile-verified
//
#include <hip/hip_runtime.h>

typedef float v2f __attribute__((ext_vector_type(2)));
typedef float v8f __attribute__((ext_vector_type(8)));

// Problem constants (from reference setup_inputs)
constexpr int Bn = 4, Hh = 128, Ww = 128, Cc = 64, Ff = 64;
constexpr int K2 = 9, CK = K2 * Cc;        // 576
constexpr int TS = 4;                      // interior tile: 4x4 = 16 pixels = WMMA M
constexpr int HT = TS + 2;                 // sampled halo 6x6 (depthwise 3x3 needs neighbors)
constexpr int XT = TS + 4;                 // x patch 8x8 (offset conv 3x3 on halo pixels)
constexpr int NOFF = HT * HT * 2 * K2;     // 648 offset values per tile
constexpr int NTAP = HT * HT * K2;         // 324 (pixel, tap) sampling tasks
constexpr int APAD = CK + 4;               // pad A rows -> conflict-free lane-strided reads

__global__ __launch_bounds__(256) void deform_fused(
    const float* __restrict__ x,    const float* __restrict__ w_off,
    const float* __restrict__ b_off,const float* __restrict__ w_dw,
    const float* __restrict__ b_dw, const float* __restrict__ w_pw,
    const float* __restrict__ b_pw, float* __restrict__ out)
{
  __shared__ float xs[XT * XT * Cc];       // 16 KB  zero-padded x patch
  __shared__ float off[HT * HT][2 * K2];   // 2.6 KB offsets for halo pixels
  __shared__ float smp[HT * HT][CK];       // 83 KB  sampled halo
  __shared__ float aT[16][APAD];           // 37 KB  depthwise output (GEMM A)
  __shared__ float swt[NTAP][4];           // bilinear corner weights
  __shared__ int   six[NTAP][4];           // bilinear corner base indices (*Cc)

  const int tid = threadIdx.x;
  const int blk = blockIdx.x;
  const int b   = blk >> 10;               // 1024 tiles per batch image
  const int rem = blk & 1023;
  const int r0  = (rem >> 5) << 2;
  const int c0  = (rem & 31) << 2;

  // ---- phase 0: stage zero-padded 8x8x64 x patch (SAME padding of offset conv) ----
  for (int i = tid; i < XT * XT * Cc; i += 256) {
    int ch = i & (Cc - 1); int p = i / Cc;
    int pc = p % XT, pr = p / XT;
    int rr = r0 - 2 + pr, cc = c0 - 2 + pc;
    float v = 0.f;
    if (rr >= 0 && rr < Hh && cc >= 0 && cc < Ww)
      v = x[((b * Hh + rr) * Ww + cc) * Cc + ch];
    xs[i] = v;
  }
  __syncthreads();

  // ---- phase 1: offset conv 3x3x64 -> 18, for all 6x6 halo pixels ----
  for (int t = tid; t < NOFF; t += 256) {
    int oc = t % (2 * K2); int p = t / (2 * K2);
    int pc = p % HT, pr = p / HT;            // halo pixel -> xs coords (pr+1, pc+1)
    float acc = b_off[oc];
    for (int kh = 0; kh < 3; ++kh)
      for (int kw = 0; kw < 3; ++kw) {
        const float* xr = &xs[((pr + kh) * XT + (pc + kw)) * Cc];
        const float* wr = &w_off[((kh * 3 + kw) * Cc) * (2 * K2) + oc];
        #pragma unroll 8
        for (int ch = 0; ch < Cc; ++ch)
          acc += xr[ch] * wr[ch * (2 * K2)];
      }
    off[p][oc] = acc;
  }
  __syncthreads();

  // ---- phase 2a: per (halo pixel, tap) bilinear weights + corner indices ----
  for (int t = tid; t < NTAP; t += 256) {
    int tap = t % K2; int p = t / K2;
    int pc = p % HT, pr = p / HT;
    int ir = r0 - 1 + pr, ic = c0 - 1 + pc;  // image coords of halo pixel
    bool inimg = (ir >= 0 && ir < Hh && ic >= 0 && ic < Ww);
    float ox = off[p][2 * tap], oy = off[p][2 * tap + 1];
    float sx = (float)ic + (float)((tap % 3) - 1) + ox;
    float sy = (float)ir + (float)((tap / 3) - 1) + oy;
    sx = fminf(fmaxf(sx, 0.f), (float)(Ww - 1));
    sy = fminf(fmaxf(sy, 0.f), (float)(Hh - 1));
    float x0 = floorf(sx), y0 = floorf(sy);
    float x1 = fminf(x0 + 1.f, (float)(Ww - 1));
    float y1 = fminf(y0 + 1.f, (float)(Hh - 1));
    float wa = (x1 - sx) * (y1 - sy);
    float wb = (x1 - sx) * (sy - y0);
    float wc = (sx - x0) * (y1 - sy);
    float wd = (sx - x0) * (sy - y0);
    if (!inimg) { wa = 0.f; wb = 0.f; wc = 0.f; wd = 0.f; } // depthwise SAME zero-pad
    int xi0 = (int)x0, xi1 = (int)x1, yi0 = (int)y0, yi1 = (int)y1;
    int base = b * Hh * Ww;
    swt[t][0] = wa; swt[t][1] = wb; swt[t][2] = wc; swt[t][3] = wd;
    six[t][0] = (base + yi0 * Ww + xi0) * Cc;
    six[t][1] = (base + yi1 * Ww + xi0) * Cc;
    six[t][2] = (base + yi0 * Ww + xi1) * Cc;
    six[t][3] = (base + yi1 * Ww + xi1) * Cc;
  }
  __syncthreads();

  // ---- phase 2b: gather 4 corners x 64 ch (coalesced over ch; L2-resident) ----
  for (int t = tid; t < NTAP * Cc; t += 256) {
    int ch = t & (Cc - 1); int pt = t / Cc;
    float v = swt[pt][0] * x[six[pt][0] + ch]
            + swt[pt][1] * x[six[pt][1] + ch]
            + swt[pt][2] * x[six[pt][2] + ch]
            + swt[pt][3] * x[six[pt][3] + ch];
    int tap = pt % K2; int p = pt / K2;
    smp[p][tap * Cc + ch] = v;
  }
  __syncthreads();

  // ---- phase 3: depthwise 3x3 over sampled halo -> GEMM A tile (16 x 576) ----
  for (int t = tid; t < 16 * CK; t += 256) {
    int j = t % CK; int p = t / CK;          // p: interior pixel, row-major 4x4
    int pr = p >> 2, pc = p & 3;
    float acc = b_dw[j];
    #pragma unroll
    for (int kh = 0; kh < 3; ++kh)
      #pragma unroll
      for (int kw = 0; kw < 3; ++kw)
        acc += smp[(pr + kh) * HT + (pc + kw)][j] * w_dw[(kh * 3 + kw) * CK + j];
    aT[p][j] = acc;
  }
  __syncthreads();

  // ---- phase 4: pointwise GEMM via V_WMMA_F32_16X16X4_F32 ----
  // M=16 pixels, N=64 split into 4 tiles over waves 0..3, K=576 in steps of 4.
  int wave = tid >> 5, lane = tid & 31;
  if (wave < 4) {
    int n0 = wave << 4;
    int m  = lane & 15;                      // A: M index / B,D: N index
    int k2 = (lane >> 4) << 1;               // lanes 0-15: K {0,1}; 16-31: K {2,3}
    v8f acc = {};
    for (int k = 0; k < CK; k += 4) {
      v2f a, bm;
      a.x  = aT[m][k + k2];
      a.y  = aT[m][k + k2 + 1];
      bm.x = w_pw[(k + k2) * Ff + n0 + m];
      bm.y = w_pw[(k + k2 + 1) * Ff + n0 + m];
      acc = __builtin_amdgcn_wmma_f32_16x16x4_f32(false, a, false, bm,
                                                  (short)0, acc, false, false);
    }
    float bias = b_pw[n0 + m];
    int mBase = (lane >> 4) << 3;            // D: VGPR j -> M=j (lanes<16) / j+8
    #pragma unroll
    for (int j = 0; j < 8; ++j) {
      int row = mBase + j;
      int pr = row >> 2, pc = row & 3;
      out[((b * Hh + (r0 + pr)) * Ww + (c0 + pc)) * Ff + n0 + m] = acc[j] + bias;
    }
  }
}

extern "C" void kernel_launch(void* const* d_in, const int* in_sizes, int n_in,
                              void* d_out, int out_size, void* d_ws, size_t ws_size,
                              hipStream_t stream) {
  (void)in_sizes; (void)n_in; (void)out_size; (void)d_ws; (void)ws_size;
  const float* x     = (const float*)d_in[0];
  const float* w_off = (const float*)d_in[1];
  const float* b_off = (const float*)d_in[2];
  const float* w_dw  = (const float*)d_in[3];
  const float* b_dw  = (const float*)d_in[4];
  const float* w_pw  = (const float*)d_in[5];
  const float* b_pw  = (const float*)d_in[6];
  float* out = (float*)d_out;

  dim3 grid(Bn * (Hh / TS) * (Ww / TS));   // 4 * 32 * 32 = 4096 tiles
  deform_fused<<<grid, 256, 0, stream>>>(x, w_off, b_off, w_dw, b_dw, w_pw, b_pw, out);
}